// HiPPOLegS_56985626083735
// MI455X (gfx1250) — compile-verified
//
#include <hip/hip_runtime.h>
#include <hip/hip_bf16.h>

#define NDEG  256     // coef dimension
#define BLOCK 256
#define ELEMS 8       // elements per thread -> 8 independent FMA chains

// -------------------------------------------------------------------------
// Pass 1: per-degree uniform constants into d_ws as float4[256]:
//   slot d (1..255): { A_d = (2d+1)/(d+1),  B_d = d/(d+1),
//                      W_{d+1} = coef[d]*sqrt(2d+3),  0 }
//   slot 0:          { 0, 0, w_1 = coef[0]*sqrt(3), 0 }
// -------------------------------------------------------------------------
__global__ void legs_const_kernel(const float* __restrict__ coef,
                                  float4* __restrict__ ws4) {
    int d = blockIdx.x * blockDim.x + threadIdx.x;
    if (d >= NDEG) return;
    float4 c;
    if (d == 0) {
        c.x = 0.0f; c.y = 0.0f;
        c.z = coef[0] * sqrtf(3.0f);
        c.w = 0.0f;
    } else {
        float df  = (float)d;
        float inv = 1.0f / (df + 1.0f);
        c.x = (2.0f * df + 1.0f) * inv;
        c.y = df * inv;
        c.z = coef[d] * sqrtf(2.0f * df + 3.0f);
        c.w = 0.0f;
    }
    ws4[d] = c;
}

// -------------------------------------------------------------------------
// TDM stage: DMA the 4096-byte constant table (global -> LDS offset 0).
// D# built per CDNA5 ISA ch.8: group0 {count=1, lds_addr, global_addr,
// type=2}; group1 {data_size=4B, tensor_dim0=1024, tensor_dim1=1,
// tile_dim0=1024, tile_dim1=1, stride0=1024}; groups 2/3 zero (2D).
// 6-arg builtin: (uint32x4, int32x8, int32x4, int32x4, int32x8, i32 cpol)
// -------------------------------------------------------------------------
#if defined(__gfx1250__) && __has_builtin(__builtin_amdgcn_tensor_load_to_lds)
#define USE_TDM 1
typedef unsigned int v4u __attribute__((ext_vector_type(4)));
typedef int          v8i __attribute__((ext_vector_type(8)));
typedef int          v4i __attribute__((ext_vector_type(4)));

__device__ __forceinline__ void tdm_stage_consts(const float4* __restrict__ gsrc,
                                                 unsigned lds_byte_off) {
    unsigned long long ga = (unsigned long long)(uintptr_t)gsrc;
    v4u g0;
    g0.x = 1u;                                   // count=1, user descriptor
    g0.y = lds_byte_off;                         // lds_addr (bytes)
    g0.z = (unsigned)(ga & 0xFFFFFFFFu);         // global_addr[31:0]
    g0.w = (unsigned)((ga >> 32) & 0x01FFFFFFu)  // global_addr[56:32]
         | (2u << 30);                           // type = 2 ("image")
    v8i g1;
    g1[0] = (2 << 16);          // data_size = 2 (4 bytes), mask/flags = 0
    g1[1] = (1024 << 16);       // tensor_dim0[15:0] = 1024  (bits 79:48)
    g1[2] = (1 << 16);          // tensor_dim0 hi = 0, tensor_dim1 = 1
    g1[3] = (1024 << 16);       // tensor_dim1 hi = 0, tile_dim0 = 1024
    g1[4] = 1;                  // tile_dim1 = 1, tile_dim2 = 0
    g1[5] = 1024;               // tensor_dim0_stride[31:0] = 1024
    g1[6] = 0;                  // stride0 hi = 0, tensor_dim1_stride lo = 0
    g1[7] = 0;
    v4i z4 = (v4i)0;            // group2 (unused, 2D tensor)
    v8i z8 = (v8i)0;            // trailing group (unused)
    __builtin_amdgcn_tensor_load_to_lds(g0, g1, z4, z4, z8, 0);
}
#else
#define USE_TDM 0
#endif

// -------------------------------------------------------------------------
// Pass 2: main recurrence.  out[j] = sum_{i=1..256} w_i P_i(2 t[j]/ct - 1)
// 4 fp32 VALU ops per (element, degree); 8-way per-thread ILP.
// -------------------------------------------------------------------------
__global__ void __launch_bounds__(BLOCK)
legs_main_kernel(const float*  __restrict__ t,
                 const float4* __restrict__ cst_g,
                 const int*    __restrict__ curr_t,
                 float*        __restrict__ out, int n) {
    __shared__ float4 cst[NDEG];      // first (only) LDS object -> offset 0

#if USE_TDM
    if (threadIdx.x == 0) {
        tdm_stage_consts(cst_g, 0u);
        __builtin_amdgcn_s_wait_tensorcnt(0);   // LDS writes complete
        // The TDM builtin carries the LDS target only as descriptor bits,
        // so LLVM cannot see that `cst` is written; without this opaque
        // potential-writer it folds all LDS reads to undef and deletes the
        // recurrence loop (observed: ds=0 in the histogram).
        asm volatile("" : : "r"(&cst[0]) : "memory");
    }
#else
    if (threadIdx.x < NDEG) cst[threadIdx.x] = cst_g[threadIdx.x];
#endif
    __syncthreads();

    float ct     = (float)curr_t[0];
    float tscale = 2.0f / ct;

    int base = blockIdx.x * (BLOCK * ELEMS) + threadIdx.x;   // coalesced

    float x[ELEMS], pm1[ELEMS], p[ELEMS], acc[ELEMS];
    float w1 = cst[0].z;
#pragma unroll
    for (int k = 0; k < ELEMS; ++k) {
        int   idx = base + k * BLOCK;
        float tv  = (idx < n) ? t[idx] : 0.0f;
        float xv  = __builtin_fmaf(tscale, tv, -1.0f);   // 2t/ct - 1
        x[k]   = xv;
        pm1[k] = 1.0f;     // P_0
        p[k]   = xv;       // P_1
        acc[k] = w1 * xv;  // w_1 * P_1
    }

    // Degree loop: P_{d+1} = A_d*x*P_d - B_d*P_{d-1};  acc += W_{d+1}*P_{d+1}
#pragma unroll 5
    for (int d = 1; d < NDEG; ++d) {
        float4 c = cst[d];            // ds_load_b128, uniform broadcast
#pragma unroll
        for (int k = 0; k < ELEMS; ++k) {
            float ax = c.x * x[k];
            float bp = c.y * pm1[k];
            float pn = __builtin_fmaf(ax, p[k], -bp);
            acc[k]   = __builtin_fmaf(c.z, pn, acc[k]);
            pm1[k]   = p[k];
            p[k]     = pn;
        }
    }

#pragma unroll
    for (int k = 0; k < ELEMS; ++k) {
        int idx = base + k * BLOCK;
        if (idx < n) out[idx] = acc[k];
    }
}

// -------------------------------------------------------------------------
extern "C" void kernel_launch(void* const* d_in, const int* in_sizes, int n_in,
                              void* d_out, int out_size, void* d_ws, size_t ws_size,
                              hipStream_t stream) {
    const float* t    = (const float*)d_in[0];
    const float* coef = (const float*)d_in[1];
    const int*   curr = (const int*)d_in[2];
    float*       out  = (float*)d_out;
    float4*      ws4  = (float4*)d_ws;      // needs 4096 B of scratch
    int n = in_sizes[0];

    legs_const_kernel<<<1, NDEG, 0, stream>>>(coef, ws4);

    int grid = (n + BLOCK * ELEMS - 1) / (BLOCK * ELEMS);
    legs_main_kernel<<<grid, BLOCK, 0, stream>>>(t, ws4, curr, out, n);
}